// EulerMisorientation3D_69380901700206
// MI455X (gfx1250) — compile-verified
//
#include <hip/hip_runtime.h>
#include <math.h>

// ---------------------------------------------------------------------------
// Euler misorientation mean, MI455X (gfx1250, wave32).
//   tr = sum_i ghat[i,i] * inv(g)[i,i];  inv(g)=g^T for rotations, and the
//   diagonal of g^T is the diagonal of g, so only g00,g11,g22 are needed
//   (no sin(Phi) required).
// Streaming-reduction: 48MB in -> 1 float out. float4 loads (global_load_b128),
// per-lane accumulation, wave reduction via V_WMMA_F32_16X16X4_F32 (ones-matrix
// trick), deterministic two-pass block reduction through d_ws.
// ---------------------------------------------------------------------------

typedef float v2f __attribute__((ext_vector_type(2)));
typedef float v8f __attribute__((ext_vector_type(8)));

#define BLOCK 256
#define GRID_MAX 1024
#define TWO_PI_F 6.28318530717958647692f
#define PI_F     3.14159265358979323846f

// Reduce 32 per-lane floats to the wave-wide sum using one WMMA.
// A(16x4): VGPR0 = v (A[m,0]=v[m] lanes0-15, A[m,2]=v[m+16] lanes16-31), VGPR1 = 0.
// B(4x16) = ones  ->  D[m,n] = v[m] + v[m+16] for every n.
// Per lane n: sum of its 8 D VGPRs = rows {0..7} (lanes 0-15) or {8..15} (16-31).
// waveSum = s(lane 0) + s(lane 16).
__device__ __forceinline__ float wave_reduce_wmma(float v) {
    v2f A; A[0] = v;    A[1] = 0.0f;
    v2f B; B[0] = 1.0f; B[1] = 1.0f;
    v8f C = {};
    v8f D = __builtin_amdgcn_wmma_f32_16x16x4_f32(
        /*neg_a=*/false, A, /*neg_b=*/false, B,
        /*c_mod=*/(short)0, C, /*reuse_a=*/false, /*reuse_b=*/false);
    float s = D[0] + D[1] + D[2] + D[3] + D[4] + D[5] + D[6] + D[7];
    return __shfl(s, 0, 32) + __shfl(s, 16, 32);
}

// Block (256 thr = 8 waves) sum; result valid on thread 0.
__device__ __forceinline__ float block_reduce(float v, float* lds8) {
    float w = wave_reduce_wmma(v);
    const int lane = threadIdx.x & 31;
    const int wave = threadIdx.x >> 5;
    if (lane == 0) lds8[wave] = w;
    __syncthreads();
    float total = 0.0f;
    if (threadIdx.x == 0) {
        #pragma unroll
        for (int i = 0; i < BLOCK / 32; ++i) total += lds8[i];
    }
    return total;
}

// Misorientation^2 for one voxel given the six angles.
__device__ __forceinline__ float voxel_mis2(float a1, float aP, float a2,
                                            float b1, float bP, float b2) {
    float s1, c1, s2, c2, t1, d1, t2, d2;
    __sincosf(a1, &s1, &c1);   // fast HW-path sin/cos (args in ~[-0.2, 6.5])
    __sincosf(a2, &s2, &c2);
    __sincosf(b1, &t1, &d1);
    __sincosf(b2, &t2, &d2);
    float cP = __cosf(aP);
    float dP = __cosf(bP);

    float gc = c1 * c2, gs = s1 * s2;
    float hc = d1 * d2, hs = t1 * t2;
    float g00 = fmaf(-gs, cP, gc);       // c1c2 - s1s2*cP
    float g11 = fmaf(gc, cP, -gs);       // c1c2*cP - s1s2
    float h00 = fmaf(-hs, dP, hc);
    float h11 = fmaf(hc, dP, -hs);

    float tr = fmaf(g00, h00, fmaf(g11, h11, cP * dP));
    float t  = 0.5f * (tr - 1.0f);
    t = fminf(1.0f, fmaxf(-1.0f, t));
    float th = acosf(t);
    return th * th;
}

__global__ void __launch_bounds__(BLOCK)
mis_partial(const float* __restrict__ x, const float* __restrict__ xh,
            float* __restrict__ ws, int N) {
    const int tid    = blockIdx.x * BLOCK + threadIdx.x;
    const int stride = gridDim.x * BLOCK;

    const float4* xa = (const float4*)(x);
    const float4* xb = (const float4*)(x + N);
    const float4* xc = (const float4*)(x + 2 * (size_t)N);
    const float4* ha = (const float4*)(xh);
    const float4* hb = (const float4*)(xh + N);
    const float4* hc4 = (const float4*)(xh + 2 * (size_t)N);

    float acc = 0.0f;
    const int nv4 = N >> 2;   // N = 128^3, divisible by 4
    for (int i = tid; i < nv4; i += stride) {
        float4 p1 = xa[i], pP = xb[i], p2 = xc[i];
        float4 q1 = ha[i], qP = hb[i], q2 = hc4[i];
        acc += voxel_mis2(p1.x * TWO_PI_F, pP.x * PI_F, p2.x * TWO_PI_F,
                          q1.x * TWO_PI_F, qP.x * PI_F, q2.x * TWO_PI_F);
        acc += voxel_mis2(p1.y * TWO_PI_F, pP.y * PI_F, p2.y * TWO_PI_F,
                          q1.y * TWO_PI_F, qP.y * PI_F, q2.y * TWO_PI_F);
        acc += voxel_mis2(p1.z * TWO_PI_F, pP.z * PI_F, p2.z * TWO_PI_F,
                          q1.z * TWO_PI_F, qP.z * PI_F, q2.z * TWO_PI_F);
        acc += voxel_mis2(p1.w * TWO_PI_F, pP.w * PI_F, p2.w * TWO_PI_F,
                          q1.w * TWO_PI_F, qP.w * PI_F, q2.w * TWO_PI_F);
    }
    // scalar tail (empty for N divisible by 4)
    for (int i = (nv4 << 2) + tid; i < N; i += stride) {
        acc += voxel_mis2(x[i] * TWO_PI_F, x[N + i] * PI_F, x[2 * (size_t)N + i] * TWO_PI_F,
                          xh[i] * TWO_PI_F, xh[N + i] * PI_F, xh[2 * (size_t)N + i] * TWO_PI_F);
    }

    __shared__ float lds8[BLOCK / 32];
    float bsum = block_reduce(acc, lds8);
    if (threadIdx.x == 0) ws[blockIdx.x] = bsum;
}

__global__ void __launch_bounds__(BLOCK)
mis_finalize(const float* __restrict__ ws, int nparts,
             float* __restrict__ out, float invN) {
    float acc = 0.0f;
    for (int i = threadIdx.x; i < nparts; i += BLOCK) acc += ws[i];
    __shared__ float lds8[BLOCK / 32];
    float total = block_reduce(acc, lds8);
    if (threadIdx.x == 0) out[0] = total * invN;
}

extern "C" void kernel_launch(void* const* d_in, const int* in_sizes, int n_in,
                              void* d_out, int out_size, void* d_ws, size_t ws_size,
                              hipStream_t stream) {
    const float* x  = (const float*)d_in[0];
    const float* xh = (const float*)d_in[1];
    float* out = (float*)d_out;
    float* ws  = (float*)d_ws;

    const int N = in_sizes[0] / 3;   // voxel count (128^3)

    int grid = GRID_MAX;
    if (ws_size < (size_t)grid * sizeof(float)) {
        grid = (int)(ws_size / sizeof(float));
        if (grid < 1) grid = 1;
    }

    mis_partial<<<grid, BLOCK, 0, stream>>>(x, xh, ws, N);
    mis_finalize<<<1, BLOCK, 0, stream>>>(ws, grid, out, 1.0f / (float)N);
}